// GCNEncoder_19928648254210
// MI455X (gfx1250) — compile-verified
//
#include <hip/hip_runtime.h>

#define NN 100000
#define NE 1600000
#define DD 128

typedef __attribute__((ext_vector_type(2))) float v2f;
typedef __attribute__((ext_vector_type(8))) float v8f;

// ---------------------------------------------------------------------------
// h[NN,DD] = x[NN,DD] @ W[DD,DD]   (fp32, exact, via V_WMMA_F32_16X16X4_F32)
// grid: (ceil(6250/8), 8), block 256 = 8 waves; wave -> one 16x16 tile of h.
// ---------------------------------------------------------------------------
__global__ __launch_bounds__(256) void gcn_gemm_wmma(const float* __restrict__ x,
                                                     const float* __restrict__ W,
                                                     float* __restrict__ h) {
    // W column-block staged transposed: ldsWT[n][k], k contiguous, pad 132
    __shared__ float ldsWT[16 * 132];
    const int tid     = threadIdx.x;
    const int colBase = blockIdx.y * 16;

    #pragma unroll
    for (int i = 0; i < 8; ++i) {
        int e = tid + i * 256;           // 0..2047
        int n = e & 15;
        int k = e >> 4;                  // 0..127
        ldsWT[n * 132 + k] = W[k * DD + colBase + n];
    }
    __syncthreads();

    const int wave  = tid >> 5;
    const int lane  = tid & 31;
    const int mTile = blockIdx.x * 8 + wave;
    if (mTile >= (NN / 16)) return;      // wave-uniform: EXEC stays all-1s

    const int    row  = mTile * 16 + (lane & 15);
    const int    koff = (lane >> 4) * 2;            // lanes 16-31 carry K=2,3
    const float* xrow = x + (size_t)row * DD;
    const float* wtn  = ldsWT + (lane & 15) * 132;

    v8f acc = {};
    #pragma unroll
    for (int kk = 0; kk < DD; kk += 4) {
        v2f a, b;
        a.x = xrow[kk + koff];
        a.y = xrow[kk + koff + 1];
        b.x = wtn[kk + koff];
        b.y = wtn[kk + koff + 1];
        acc = __builtin_amdgcn_wmma_f32_16x16x4_f32(
                  /*neg_a=*/false, a, /*neg_b=*/false, b,
                  /*c_mod=*/(short)0, acc, /*reuse_a=*/false, /*reuse_b=*/false);
    }

    // D layout: VGPR v -> row (mTile*16 + (lane>>4)*8 + v), col = lane&15
    const int col     = colBase + (lane & 15);
    const int rowBase = mTile * 16 + (lane >> 4) * 8;
    #pragma unroll
    for (int v = 0; v < 8; ++v) {
        h[(size_t)(rowBase + v) * DD + col] = acc[v];
    }
}

// agg[n][d] = bias[d]  (fold bias into the accumulation buffer)
__global__ __launch_bounds__(256) void gcn_init_bias(float* __restrict__ agg,
                                                     const float* __restrict__ b) {
    size_t i = (size_t)blockIdx.x * 256 + threadIdx.x;
    agg[i] = b[i & (DD - 1)];
}

// one wave per edge; lane handles 4 consecutive floats; L2-resident atomics
__global__ __launch_bounds__(256) void gcn_scatter(const float* __restrict__ h,
                                                   const int* __restrict__ src,
                                                   const int* __restrict__ dst,
                                                   const float* __restrict__ w,
                                                   float* __restrict__ agg) {
    const int e = blockIdx.x * 8 + (threadIdx.x >> 5);
    if (e >= NE) return;
    const int   lane = threadIdx.x & 31;
    const int   s    = src[e];
    const int   d    = dst[e];
    const float wt   = w[e];

    const float4 hv = *(const float4*)(h + (size_t)s * DD + lane * 4);
    float* out = agg + (size_t)d * DD + lane * 4;
    unsafeAtomicAdd(out + 0, hv.x * wt);
    unsafeAtomicAdd(out + 1, hv.y * wt);
    unsafeAtomicAdd(out + 2, hv.z * wt);
    unsafeAtomicAdd(out + 3, hv.w * wt);
}

// out = relu(in), elementwise (in may alias out)
__global__ __launch_bounds__(256) void gcn_relu(const float* __restrict__ in,
                                                float* __restrict__ out) {
    size_t i = (size_t)blockIdx.x * 256 + threadIdx.x;
    out[i] = fmaxf(in[i], 0.0f);
}

extern "C" void kernel_launch(void* const* d_in, const int* in_sizes, int n_in,
                              void* d_out, int out_size, void* d_ws, size_t ws_size,
                              hipStream_t stream) {
    const float* x    = (const float*)d_in[0];
    const int*   esrc = (const int*)d_in[1];
    const int*   edst = (const int*)d_in[2];
    const float* ew   = (const float*)d_in[3];
    const float* W    = (const float*)d_in[4];
    const float* b    = (const float*)d_in[5];
    float*       out  = (float*)d_out;

    float* ws0 = (float*)d_ws;                    // h scratch  (51.2 MB)
    float* ws1 = ws0 + (size_t)NN * DD;           // agg / x ping (51.2 MB)

    // layer buffer schedule (x_cur, agg, relu_dst):
    //   L0: x=in,  agg=ws1, relu->ws1
    //   L1: x=ws1, agg=out, relu->out
    //   L2: x=out, agg=ws1, relu->out
    const float* xcur[3] = { x,   ws1, out };
    float*       agg[3]  = { ws1, out, ws1 };
    float*       post[3] = { ws1, out, out };

    const dim3 gGemm((NN / 16 + 7) / 8, DD / 16); // (782, 8)
    const int  gND = (NN * DD) / 256;             // 50000
    const int  gE  = NE / 8;                      // 200000

    for (int l = 0; l < 3; ++l) {
        gcn_gemm_wmma<<<gGemm, 256, 0, stream>>>(xcur[l], W + (size_t)l * DD * DD, ws0);
        gcn_init_bias<<<gND, 256, 0, stream>>>(agg[l], b + (size_t)l * DD);
        gcn_scatter<<<gE, 256, 0, stream>>>(ws0, esrc, edst, ew, agg[l]);
        gcn_relu<<<gND, 256, 0, stream>>>(agg[l], post[l]);
    }
}